// FrequencyAdaptiveNorm_52630529245619
// MI455X (gfx1250) — compile-verified
//
#include <hip/hip_runtime.h>
#include <math.h>

// FrequencyAdaptiveNorm for MI455X (gfx1250).
// x:(32,4096,256) f32 -> per-row LayerNorm over F=256, scaled by a per-timestep
// class table (4 classes from the nested window masks). Memory-bound:
// 256 MiB traffic -> ~11us at 23.3 TB/s. WMMA f32 16x16x4 does the cross-lane
// row-sum reduction; async global->LDS b128 stages the tile.

#define S_LEN   4096
#define F_DIM   256
#define LN_EPS  1e-5f
#define PAD     260          // dwords per row in LDS (256 + 4 -> bank spread)
#define WROWS   16           // rows per wave (one WMMA covers 16 rows)
#define WAVES   2            // waves per block
#define BLOCK   (WAVES * 32)

typedef __attribute__((ext_vector_type(2))) float v2f;
typedef __attribute__((ext_vector_type(8))) float v8f;
typedef int v4i __attribute__((vector_size(16)));
typedef __attribute__((address_space(1))) v4i* gas_v4i;   // global int4*
typedef __attribute__((address_space(3))) v4i* las_v4i;   // LDS int4*

#if __has_builtin(__builtin_amdgcn_wmma_f32_16x16x4_f32)
#define HAVE_WMMA_F32X4 1
#else
#define HAVE_WMMA_F32X4 0
#endif
#if __has_builtin(__builtin_amdgcn_global_load_async_to_lds_b128)
#define HAVE_ASYNC_LDS 1
#else
#define HAVE_ASYNC_LDS 0
#endif

// ---- prep: softmax(weights) + 4-class geff/beff table (4*2*256 f32 = 8KB) ----
__global__ __launch_bounds__(256)
void fan_prep(const float* __restrict__ gamma, const float* __restrict__ beta,
              const float* __restrict__ wlog, float* __restrict__ tab) {
  const int f = threadIdx.x;  // 0..255
  const float w0 = wlog[0], w1 = wlog[1], w2 = wlog[2];
  const float mx = fmaxf(w0, fmaxf(w1, w2));
  const float e0 = expf(w0 - mx), e1 = expf(w1 - mx), e2 = expf(w2 - mx);
  const float inv = 1.0f / (e0 + e1 + e2);
  const float w[3] = {e0 * inv, e1 * inv, e2 * inv};
  float G = 0.0f, Bv = 0.0f;
  for (int cls = 0; cls < 4; ++cls) {
    tab[cls * 2 * F_DIM + f] = G;
    tab[cls * 2 * F_DIM + F_DIM + f] = Bv;
    if (cls < 3) {
      G  += w[cls] * gamma[cls * F_DIM + f];
      Bv += w[cls] * beta[cls * F_DIM + f];
    }
  }
}

// ---- main: per-wave 16-row LN tile ----
__global__ __launch_bounds__(BLOCK)
void fan_main(const float* __restrict__ x, const float* __restrict__ tab,
              float* __restrict__ out) {
  __shared__ float lds[WAVES * WROWS * PAD];
  __shared__ float stats[WAVES * 2 * WROWS];  // per wave: 16 means, 16 rstds

  const int lane = threadIdx.x & 31;
  const int wv   = threadIdx.x >> 5;
  float* wlds = &lds[wv * WROWS * PAD];
  float* wst  = &stats[wv * 2 * WROWS];

  const size_t row0 = ((size_t)blockIdx.x * WAVES + wv) * WROWS;
  const float4* src4 = (const float4*)(x + row0 * F_DIM);
  float4* dst4 = (float4*)(out + row0 * F_DIM);

  // ---- stage 1: 16 rows x 256 f32 (16KB) global -> LDS, coalesced b128
#pragma unroll
  for (int j = 0; j < 32; ++j) {
    const int p  = lane + j * 32;   // float4 index in row-major tile
    const int r  = p >> 6;          // 64 float4 per row
    const int c4 = p & 63;
    float* lp = &wlds[r * PAD + c4 * 4];
#if HAVE_ASYNC_LDS
    __builtin_amdgcn_global_load_async_to_lds_b128(
        (gas_v4i)(src4 + p), (las_v4i)lp, 0, 0);
#else
    *(float4*)lp = src4[p];
#endif
  }
#if HAVE_ASYNC_LDS
#if __has_builtin(__builtin_amdgcn_s_wait_asynccnt)
  __builtin_amdgcn_s_wait_asynccnt(0);
#else
  asm volatile("s_wait_asynccnt 0" ::: "memory");
#endif
#endif
  __syncthreads();

  // ---- stage 2: per-lane 64-wide partials laid out as the 16x4 f32 A matrix,
  //               one WMMA vs all-ones B => all 16 row sums (and sum of squares)
  {
    const int r  = lane & 15;       // row this lane contributes to
    const int hi = lane >> 4;       // 0: cols 0..127 (K0,K1); 1: cols 128..255 (K2,K3)
    const float* rp = &wlds[r * PAD + hi * 128];
    float pa0 = 0.f, pa1 = 0.f, qa0 = 0.f, qa1 = 0.f;
#pragma unroll
    for (int k = 0; k < 16; ++k) {
      const float4 v0 = *(const float4*)(rp + k * 4);
      const float4 v1 = *(const float4*)(rp + 64 + k * 4);
      pa0 += v0.x + v0.y + v0.z + v0.w;
      qa0 += v0.x * v0.x + v0.y * v0.y + v0.z * v0.z + v0.w * v0.w;
      pa1 += v1.x + v1.y + v1.z + v1.w;
      qa1 += v1.x * v1.x + v1.y * v1.y + v1.z * v1.z + v1.w * v1.w;
    }
#if HAVE_WMMA_F32X4
    v2f A;    A.x = pa0;  A.y = pa1;
    v2f Q;    Q.x = qa0;  Q.y = qa1;
    v2f Ones; Ones.x = 1.0f; Ones.y = 1.0f;
    v8f Z = {0.f, 0.f, 0.f, 0.f, 0.f, 0.f, 0.f, 0.f};
    // D[m][n] = sum_k A[m][k] -> every column holds the row sum
    v8f sum = __builtin_amdgcn_wmma_f32_16x16x4_f32(
        false, A, false, Ones, (short)0, Z, false, false);
    v8f ssq = __builtin_amdgcn_wmma_f32_16x16x4_f32(
        false, Q, false, Ones, (short)0, Z, false, false);
    if ((lane & 15) == 0) {   // lane 0 -> rows 0..7, lane 16 -> rows 8..15
#pragma unroll
      for (int v = 0; v < 8; ++v) {
        const float mean = sum[v] * (1.0f / F_DIM);
        const float var  = ssq[v] * (1.0f / F_DIM) - mean * mean;
        wst[hi * 8 + v] = mean;
        wst[WROWS + hi * 8 + v] = 1.0f / sqrtf(var + LN_EPS);
      }
    }
#else
    float t1 = pa0 + pa1, t2 = qa0 + qa1;
    t1 += __shfl_xor(t1, 16, 32);
    t2 += __shfl_xor(t2, 16, 32);
    if (lane < 16) {
      const float mean = t1 * (1.0f / F_DIM);
      const float var  = t2 * (1.0f / F_DIM) - mean * mean;
      wst[r] = mean;
      wst[WROWS + r] = 1.0f / sqrtf(var + LN_EPS);
    }
#endif
  }
  __syncthreads();

  // ---- stage 3: normalize + class-table scale, coalesced b128 stores
  const int sbase = (int)(row0 & (S_LEN - 1));  // tiles never cross a batch edge
#pragma unroll 4
  for (int j = 0; j < 32; ++j) {
    const int p  = lane + j * 32;
    const int r  = p >> 6;
    const int c4 = p & 63;
    const float4 v = *(const float4*)&wlds[r * PAD + c4 * 4];
    const float mean = wst[r];
    const float rstd = wst[WROWS + r];
    const int s = sbase + r;
    const int cls = (int)(s >= 2 && s < 4094) + (int)(s >= 5 && s < 4092) +
                    (int)(s >= 10 && s < 4087);
    const float4 g4 = *(const float4*)&tab[cls * 2 * F_DIM + c4 * 4];
    const float4 b4 = *(const float4*)&tab[cls * 2 * F_DIM + F_DIM + c4 * 4];
    float4 o;
    o.x = (v.x - mean) * rstd * g4.x + b4.x;
    o.y = (v.y - mean) * rstd * g4.y + b4.y;
    o.z = (v.z - mean) * rstd * g4.z + b4.z;
    o.w = (v.w - mean) * rstd * g4.w + b4.w;
    dst4[p] = o;
  }
}

extern "C" void kernel_launch(void* const* d_in, const int* in_sizes, int n_in,
                              void* d_out, int out_size, void* d_ws, size_t ws_size,
                              hipStream_t stream) {
  const float* x     = (const float*)d_in[0];
  const float* gamma = (const float*)d_in[1];
  const float* beta  = (const float*)d_in[2];
  const float* wlog  = (const float*)d_in[3];
  float* tab = (float*)d_ws;                 // 8 KB class table
  float* o   = (float*)d_out;

  const int rows   = in_sizes[0] / F_DIM;            // 131072
  const int blocks = rows / (WAVES * WROWS);         // 4096

  fan_prep<<<1, F_DIM, 0, stream>>>(gamma, beta, wlog, tab);
  fan_main<<<blocks, BLOCK, 0, stream>>>(x, tab, o);
}